// DNCBody_61297773249163
// MI455X (gfx1250) — compile-verified
//
#include <hip/hip_runtime.h>
#include <hip/hip_bf16.h>
#include <math.h>

typedef float v2f __attribute__((ext_vector_type(2)));
typedef float v8f __attribute__((ext_vector_type(8)));
typedef int   v4i __attribute__((vector_size(16)));

#define EPSF 1e-8f

#if defined(__has_builtin)
#if __has_builtin(__builtin_amdgcn_global_load_async_to_lds_b128)
#define HAVE_ASYNC_LDS 1
#endif
#endif
#ifndef HAVE_ASYNC_LDS
#define HAVE_ASYNC_LDS 0
#endif

__device__ __forceinline__ float sigm(float x) { return 1.0f / (1.0f + expf(-x)); }
__device__ __forceinline__ float softplusf(float x) { return x > 20.0f ? x : log1pf(expf(x)); }
__device__ __forceinline__ int mini(int a, int b) { return a < b ? a : b; }

#if HAVE_ASYNC_LDS
// async 16B global -> LDS copy (GLOBAL_LOAD_ASYNC_TO_LDS_B128, ASYNCcnt)
__device__ __forceinline__ void async_cp16(const float* g, float* l) {
  __builtin_amdgcn_global_load_async_to_lds_b128(
      (__attribute__((address_space(1))) v4i*)g,
      (__attribute__((address_space(3))) v4i*)l, 0, 0);
}
__device__ __forceinline__ void wait_async0() {
#if __has_builtin(__builtin_amdgcn_s_wait_asynccnt)
  __builtin_amdgcn_s_wait_asynccnt(0);
#else
  asm volatile("s_wait_asynccnt 0x0" ::: "memory");
#endif
}
__device__ __forceinline__ void wait_async1() {
#if __has_builtin(__builtin_amdgcn_s_wait_asynccnt)
  __builtin_amdgcn_s_wait_asynccnt(1);
#else
  asm volatile("s_wait_asynccnt 0x1" ::: "memory");
#endif
}
__device__ __forceinline__ void wait_async4() {
#if __has_builtin(__builtin_amdgcn_s_wait_asynccnt)
  __builtin_amdgcn_s_wait_asynccnt(4);
#else
  asm volatile("s_wait_asynccnt 0x4" ::: "memory");
#endif
}
#endif

// ---------------------------------------------------------------------------
// C[M,N] (ldc) = A[M,K] @ W[N,K]^T  (+bias1 +bias2) (+C if acc)
// 256 threads = 8 wave32s; WG tile 64x64; each wave: 16 rows x 32 cols
// (two v8f accumulators sharing one A fragment) via V_WMMA_F32_16X16X4_F32.
// B-fragment {W[n][k],W[n][k+1]} is contiguous in row-major W, so both A and
// W are staged un-transposed. Double-buffered LDS; async G->LDS if available.
// Requires M%64==0, K%32==0. N ragged: OOB rows clamped (cols masked at store).
// ---------------------------------------------------------------------------
__global__ __launch_bounds__(256) void gemm_xwT_wmma(
    const float* __restrict__ A, const float* __restrict__ W,
    const float* __restrict__ bias1, const float* __restrict__ bias2,
    float* __restrict__ C, int M, int N, int K, int ldc, int acc)
{
  __shared__ float As[2][64][36];   // [buf][m][k], row = 144B (16B aligned)
  __shared__ float Wn[2][64][36];   // [buf][n][k]

  const int tid  = threadIdx.x;
  const int wave = tid >> 5;
  const int lane = tid & 31;
  const int wm   = wave >> 1;          // 0..3  -> rows wm*16
  const int wn0  = (wave & 1) * 32;    // 0/32  -> cols wn0, wn0+16
  const int m_blk = blockIdx.y * 64;
  const int n_blk = blockIdx.x * 64;

  const int half = lane >> 4;          // 0/1
  const int lm   = lane & 15;

  const int r0 = tid >> 3;             // 0..31 (loader row)
  const int kk0 = (tid & 7) << 2;      // 0,4,..,28 (loader k)

  v8f acc0 = {0.f,0.f,0.f,0.f,0.f,0.f,0.f,0.f};
  v8f acc1 = {0.f,0.f,0.f,0.f,0.f,0.f,0.f,0.f};

  int buf = 0;

#if HAVE_ASYNC_LDS
  // ---- async double-buffered pipeline ----
  {
    int g0 = mini(n_blk + r0, N - 1), g1 = mini(n_blk + r0 + 32, N - 1);
    async_cp16(&A[(size_t)(m_blk + r0) * K + kk0],      &As[0][r0][kk0]);
    async_cp16(&A[(size_t)(m_blk + r0 + 32) * K + kk0], &As[0][r0 + 32][kk0]);
    async_cp16(&W[(size_t)g0 * K + kk0],                &Wn[0][r0][kk0]);
    async_cp16(&W[(size_t)g1 * K + kk0],                &Wn[0][r0 + 32][kk0]);
  }
  for (int k0 = 0; k0 < K; k0 += 32) {
    if (k0 + 32 < K) {
      int nb = buf ^ 1, kn = k0 + 32;
      int g0 = mini(n_blk + r0, N - 1), g1 = mini(n_blk + r0 + 32, N - 1);
      async_cp16(&A[(size_t)(m_blk + r0) * K + kn + kk0],      &As[nb][r0][kk0]);
      async_cp16(&A[(size_t)(m_blk + r0 + 32) * K + kn + kk0], &As[nb][r0 + 32][kk0]);
      async_cp16(&W[(size_t)g0 * K + kn + kk0],                &Wn[nb][r0][kk0]);
      async_cp16(&W[(size_t)g1 * K + kn + kk0],                &Wn[nb][r0 + 32][kk0]);
      wait_async4();                 // current tile (first 4 ops) landed
    } else {
      wait_async0();
    }
    __syncthreads();
    #pragma unroll
    for (int kk = 0; kk < 32; kk += 4) {
      v2f a  = *(const v2f*)&As[buf][wm * 16 + lm][kk + half * 2];
      v2f b0 = *(const v2f*)&Wn[buf][wn0 + lm][kk + half * 2];
      v2f b1 = *(const v2f*)&Wn[buf][wn0 + 16 + lm][kk + half * 2];
      acc0 = __builtin_amdgcn_wmma_f32_16x16x4_f32(false, a, false, b0, (short)0, acc0, false, false);
      acc1 = __builtin_amdgcn_wmma_f32_16x16x4_f32(false, a, false, b1, (short)0, acc1, false, false);
    }
    __syncthreads();
    buf ^= 1;
  }
#else
  // ---- register-staged double-buffered pipeline ----
  float4 aR0, aR1, wR0, wR1;
  {
    int g0 = mini(n_blk + r0, N - 1), g1 = mini(n_blk + r0 + 32, N - 1);
    aR0 = *(const float4*)&A[(size_t)(m_blk + r0) * K + kk0];
    aR1 = *(const float4*)&A[(size_t)(m_blk + r0 + 32) * K + kk0];
    wR0 = *(const float4*)&W[(size_t)g0 * K + kk0];
    wR1 = *(const float4*)&W[(size_t)g1 * K + kk0];
    *(float4*)&As[0][r0][kk0] = aR0; *(float4*)&As[0][r0 + 32][kk0] = aR1;
    *(float4*)&Wn[0][r0][kk0] = wR0; *(float4*)&Wn[0][r0 + 32][kk0] = wR1;
  }
  __syncthreads();
  for (int k0 = 32; k0 < K; k0 += 32) {
    int g0 = mini(n_blk + r0, N - 1), g1 = mini(n_blk + r0 + 32, N - 1);
    aR0 = *(const float4*)&A[(size_t)(m_blk + r0) * K + k0 + kk0];
    aR1 = *(const float4*)&A[(size_t)(m_blk + r0 + 32) * K + k0 + kk0];
    wR0 = *(const float4*)&W[(size_t)g0 * K + k0 + kk0];
    wR1 = *(const float4*)&W[(size_t)g1 * K + k0 + kk0];
    #pragma unroll
    for (int kk = 0; kk < 32; kk += 4) {
      v2f a  = *(const v2f*)&As[buf][wm * 16 + lm][kk + half * 2];
      v2f b0 = *(const v2f*)&Wn[buf][wn0 + lm][kk + half * 2];
      v2f b1 = *(const v2f*)&Wn[buf][wn0 + 16 + lm][kk + half * 2];
      acc0 = __builtin_amdgcn_wmma_f32_16x16x4_f32(false, a, false, b0, (short)0, acc0, false, false);
      acc1 = __builtin_amdgcn_wmma_f32_16x16x4_f32(false, a, false, b1, (short)0, acc1, false, false);
    }
    int nb = buf ^ 1;
    *(float4*)&As[nb][r0][kk0] = aR0; *(float4*)&As[nb][r0 + 32][kk0] = aR1;
    *(float4*)&Wn[nb][r0][kk0] = wR0; *(float4*)&Wn[nb][r0 + 32][kk0] = wR1;
    __syncthreads();
    buf ^= 1;
  }
  #pragma unroll
  for (int kk = 0; kk < 32; kk += 4) {
    v2f a  = *(const v2f*)&As[buf][wm * 16 + lm][kk + half * 2];
    v2f b0 = *(const v2f*)&Wn[buf][wn0 + lm][kk + half * 2];
    v2f b1 = *(const v2f*)&Wn[buf][wn0 + 16 + lm][kk + half * 2];
    acc0 = __builtin_amdgcn_wmma_f32_16x16x4_f32(false, a, false, b0, (short)0, acc0, false, false);
    acc1 = __builtin_amdgcn_wmma_f32_16x16x4_f32(false, a, false, b1, (short)0, acc1, false, false);
  }
#endif

  // ---- epilogue: C/D layout row = v + half*8, col = lm ----
  #pragma unroll
  for (int t = 0; t < 2; t++) {
    int col = n_blk + wn0 + t * 16 + lm;
    if (col < N) {
      float badd = 0.f;
      if (bias1) badd += bias1[col];
      if (bias2) badd += bias2[col];
      v8f accv = t ? acc1 : acc0;
      #pragma unroll
      for (int v = 0; v < 8; v++) {
        int row = m_blk + wm * 16 + v + half * 8;
        float val = accv[v] + badd;
        float* p = &C[(size_t)row * ldc + col];
        if (acc) val += *p;
        *p = val;
      }
    }
  }
}

// ---------------------------------------------------------------------------
__global__ __launch_bounds__(256) void lstm_pointwise(
    const float* __restrict__ gates, const float* __restrict__ c,
    float* __restrict__ nh, int BH, int H)
{
  int idx = blockIdx.x * 256 + threadIdx.x;
  if (idx >= BH) return;
  int bb = idx / H, j = idx % H;
  const float* g = gates + (size_t)bb * 4 * H;
  float gi = g[j], gf = g[H + j], gg = g[2 * H + j], go = g[3 * H + j];
  float nc = sigm(gf) * c[idx] + sigm(gi) * tanhf(gg);
  nh[idx] = sigm(go) * tanhf(nc);
}

// ---------------------------------------------------------------------------
// Fused DNC tail: one workgroup per batch element.
// ---------------------------------------------------------------------------
#define Nn 128
#define Mm 64
#define Rr 4
#define HEADC 471
#define LCH 8          // link chunk rows
#define NCH (Nn / LCH)

__global__ __launch_bounds__(256) void dnc_tail(
    const float* __restrict__ ctrl,    // B x 471
    const float* __restrict__ memory,  // B x N x M
    const float* __restrict__ usage,   // B x N
    const float* __restrict__ link,    // B x N x N
    const float* __restrict__ prec,    // B x N
    const float* __restrict__ prw,     // B x R x N
    const float* __restrict__ pww,     // B x 1 x N
    float* __restrict__ reads_out)     // B x (R*M)
{
  const int b   = blockIdx.x;
  const int tid = threadIdx.x;

  __shared__ float memS[Nn][Mm];        // memory, then mem_new (in place)
  __shared__ float linkS[2][LCH][132];  // double-buffered link chunk
  __shared__ float prwS[Rr][Nn];
  __shared__ float fwdS[Rr][Nn];
  __shared__ float bwdS[Rr][Nn];
  __shared__ float crS[Rr][Nn];         // cr scores -> wr (in place)
  __shared__ float krS[Rr][Mm];
  __shared__ float kwS[Mm], eraseS[Mm], addvS[Mm];
  __shared__ float svS[Nn], allocS[Nn], cwS[Nn], wwS[Nn], precS[Nn], dotS[Nn], mnS[Nn];
  __shared__ int   siS[Nn];
  __shared__ float red[Nn];
  __shared__ float betarS[Rr], fgS[Rr], piS[Rr][3], piRaw[Rr * 3];
  __shared__ float krnS[Rr], mxS[Rr], smS[Rr];
  __shared__ float scal[4];             // 0:betaw 1:ga 2:gw 3:kwn

  // ---- Phase A: stage memory (async, overlapped) + ctrl nonlinearities ----
  {
    const float* msrc = memory + (size_t)b * Nn * Mm;
#if HAVE_ASYNC_LDS
    float* mdstf = &memS[0][0];
    for (int i = tid; i < (Nn * Mm) / 4; i += 256)
      async_cp16(msrc + i * 4, mdstf + i * 4);
#endif
    const float* cb = ctrl + (size_t)b * HEADC;
    for (int i = tid; i < HEADC; i += 256) {
      float x = cb[i];
      if      (i < 256) krS[i >> 6][i & 63] = tanhf(x);
      else if (i < 260) betarS[i - 256] = softplusf(x);
      else if (i < 324) kwS[i - 260] = tanhf(x);
      else if (i < 325) scal[0] = softplusf(x);
      else if (i < 389) eraseS[i - 325] = sigm(x);
      else if (i < 453) addvS[i - 389] = tanhf(x);
      else if (i < 454) scal[1] = sigm(x);
      else if (i < 455) scal[2] = sigm(x);
      else if (i < 459) fgS[i - 455] = sigm(x);
      else              piRaw[i - 459] = x;
    }
#if !HAVE_ASYNC_LDS
    const float4* msrc4 = (const float4*)msrc;
    float4* mdst = (float4*)&memS[0][0];
    for (int i = tid; i < (Nn * Mm) / 4; i += 256) mdst[i] = msrc4[i];
#endif
    for (int i = tid; i < Rr * Nn; i += 256)
      prwS[i >> 7][i & 127] = prw[(size_t)b * Rr * Nn + i];
    if (tid < Nn) precS[tid] = prec[(size_t)b * Nn + tid];
  }
#if HAVE_ASYNC_LDS
  wait_async0();
#endif
  __syncthreads();

  // ---- Phase B: usage u, memory row norms, pi softmax, key norms ----
  if (tid < Nn) {
    float us = usage[(size_t)b * Nn + tid];
    float pw = pww[(size_t)b * Nn + tid];
    float psi = 1.0f;
    #pragma unroll
    for (int r = 0; r < Rr; r++) psi *= (1.0f - fgS[r] * prwS[r][tid]);
    float pu = 0.005f + 0.995f * us;
    float u  = (pu + (1.0f - pu) * pw) * psi;   // reg_ww == prev_write_w
    svS[tid] = u; siS[tid] = tid;
    float s2 = 0.f;
    #pragma unroll
    for (int m = 0; m < Mm; m++) { float v = memS[tid][m]; s2 += v * v; }
    mnS[tid] = fmaxf(sqrtf(s2), EPSF);
  }
  if (tid >= 128 && tid < 128 + Rr) {           // pi softmax (per r, over 3)
    int r = tid - 128;
    float a0 = piRaw[3*r], a1 = piRaw[3*r+1], a2 = piRaw[3*r+2];
    float mx = fmaxf(a0, fmaxf(a1, a2));
    float e0 = expf(a0 - mx), e1 = expf(a1 - mx), e2 = expf(a2 - mx);
    float inv = 1.0f / (e0 + e1 + e2);
    piS[r][0] = e0 * inv; piS[r][1] = e1 * inv; piS[r][2] = e2 * inv;
  }
  if (tid >= 136 && tid < 136 + Rr) {           // read-key norms
    int r = tid - 136; float s2 = 0.f;
    #pragma unroll
    for (int m = 0; m < Mm; m++) { float v = krS[r][m]; s2 += v * v; }
    krnS[r] = fmaxf(sqrtf(s2), EPSF);
  }
  if (tid == 140) {                              // write-key norm
    float s2 = 0.f;
    #pragma unroll
    for (int m = 0; m < Mm; m++) { float v = kwS[m]; s2 += v * v; }
    scal[3] = fmaxf(sqrtf(s2), EPSF);
  }
  __syncthreads();

  // ---- Phase C: bitonic sort (ascending) of u with payload index ----
  for (int k = 2; k <= Nn; k <<= 1) {
    for (int j = k >> 1; j > 0; j >>= 1) {
      if (tid < Nn) {
        int ixj = tid ^ j;
        if (ixj > tid) {
          bool up = ((tid & k) == 0);
          float v0 = svS[tid], v1 = svS[ixj];
          if (up ? (v0 > v1) : (v0 < v1)) {
            svS[tid] = v1; svS[ixj] = v0;
            int t = siS[tid]; siS[tid] = siS[ixj]; siS[ixj] = t;
          }
        }
      }
      __syncthreads();
    }
  }

  // ---- Phase D: write-content scores; serial excl-cumprod allocation ----
  if (tid < Nn) {
    float dot = 0.f;
    #pragma unroll
    for (int m = 0; m < Mm; m++) dot += kwS[m] * memS[tid][m];
    dotS[tid] = scal[0] * (dot / (scal[3] * mnS[tid]));
  }
  if (tid == 255) {
    float p = 1.0f;
    for (int i = 0; i < Nn; i++) {
      allocS[siS[i]] = (1.0f - svS[i]) * p;
      p *= svS[i];
    }
  }
  __syncthreads();

  // ---- Phase E: softmax(cw), ww ----
  if (tid < Nn) red[tid] = dotS[tid];
  __syncthreads();
  for (int s = 64; s >= 1; s >>= 1) {
    if (tid < s) red[tid] = fmaxf(red[tid], red[tid + s]);
    __syncthreads();
  }
  float cwmax = red[0];
  __syncthreads();
  if (tid < Nn) { float e = expf(dotS[tid] - cwmax); dotS[tid] = e; red[tid] = e; }
  __syncthreads();
  for (int s = 64; s >= 1; s >>= 1) {
    if (tid < s) red[tid] += red[tid + s];
    __syncthreads();
  }
  float cwinv = 1.0f / red[0];
  __syncthreads();
  if (tid < Nn) {
    float cw = dotS[tid] * cwinv;
    cwS[tid] = cw;
    wwS[tid] = scal[2] * (scal[1] * allocS[tid] + (1.0f - scal[1]) * cw);
  }
  __syncthreads();

  // ---- Phase F: mem_new in place; zero fwd/bwd ----
  for (int i = tid; i < Nn * Mm; i += 256) {
    int n = i >> 6, m = i & 63;
    memS[n][m] = memS[n][m] * (1.0f - wwS[n] * eraseS[m]) + wwS[n] * addvS[m];
  }
  for (int i = tid; i < Rr * Nn; i += 256) {
    (&fwdS[0][0])[i] = 0.f; (&bwdS[0][0])[i] = 0.f;
  }
  __syncthreads();

  // ---- Phase G: stream link in 8-row chunks, fuse Lnew into fwd/bwd ----
  const float* lbase = link + (size_t)b * Nn * Nn;
  int lbuf = 0;
  {
    const int lr = tid >> 5, lc4 = (tid & 31) << 2;  // 8 rows x 32 float4
#if HAVE_ASYNC_LDS
    async_cp16(&lbase[(size_t)lr * Nn + lc4], &linkS[0][lr][lc4]);
#endif
    for (int cix = 0; cix < NCH; cix++) {
      int j0 = cix * LCH;
#if HAVE_ASYNC_LDS
      if (cix + 1 < NCH) {
        async_cp16(&lbase[(size_t)(j0 + LCH + lr) * Nn + lc4],
                   &linkS[lbuf ^ 1][lr][lc4]);
        wait_async1();                  // current chunk landed
      } else {
        wait_async0();
      }
#else
      *(float4*)&linkS[lbuf][lr][lc4] =
          *(const float4*)&lbase[(size_t)(j0 + lr) * Nn + lc4];
      if (cix + 1 < NCH && tid < LCH)
        __builtin_prefetch(&lbase[(size_t)(j0 + LCH + tid) * Nn], 0, 0);
#endif
      __syncthreads();

      if (tid < 128) {
        // fwd: 16 threads per chunk row, 8 columns each; combine via LDS atomics
        int row = j0 + (tid >> 4);
        int jsg = (tid & 15) << 3;
        float wi = wwS[row], facc[Rr] = {0.f, 0.f, 0.f, 0.f};
        #pragma unroll
        for (int jj = 0; jj < 8; jj++) {
          int j = jsg + jj;
          float l = (row == j) ? 0.f
                  : (1.0f - wi - wwS[j]) * linkS[lbuf][row - j0][j] + wi * precS[j];
          #pragma unroll
          for (int r = 0; r < Rr; r++) facc[r] += l * prwS[r][j];
        }
        #pragma unroll
        for (int r = 0; r < Rr; r++) atomicAdd(&fwdS[r][row], facc[r]);
      } else {
        // bwd: thread owns column i, accumulates this chunk's 8 rows
        int i = tid - 128;
        float wiC = wwS[i], bacc[Rr] = {0.f, 0.f, 0.f, 0.f};
        #pragma unroll
        for (int rr = 0; rr < LCH; rr++) {
          int jrow = j0 + rr;
          float wj = wwS[jrow];
          float l = (jrow == i) ? 0.f
                  : (1.0f - wj - wiC) * linkS[lbuf][rr][i] + wj * precS[i];
          #pragma unroll
          for (int r = 0; r < Rr; r++) bacc[r] += l * prwS[r][jrow];
        }
        #pragma unroll
        for (int r = 0; r < Rr; r++) bwdS[r][i] += bacc[r];
      }
      __syncthreads();
      lbuf ^= 1;
    }
  }

  // ---- Phase H: mem_new row norms ----
  if (tid < Nn) {
    float s2 = 0.f;
    #pragma unroll
    for (int m = 0; m < Mm; m++) { float v = memS[tid][m]; s2 += v * v; }
    mnS[tid] = fmaxf(sqrtf(s2), EPSF);
  }
  __syncthreads();

  // ---- Phase I: read-content softmax + read weights ----
  for (int p = 0; p < 2; p++) {
    int idx = tid + p * 256;
    int r = idx >> 7, n = idx & 127;
    float dot = 0.f;
    #pragma unroll
    for (int m = 0; m < Mm; m++) dot += krS[r][m] * memS[n][m];
    crS[r][n] = betarS[r] * (dot / (krnS[r] * mnS[n]));
  }
  __syncthreads();
  if (tid < Rr) {
    float mx = -1e30f;
    for (int n = 0; n < Nn; n++) mx = fmaxf(mx, crS[tid][n]);
    mxS[tid] = mx;
  }
  __syncthreads();
  for (int p = 0; p < 2; p++) {
    int idx = tid + p * 256;
    int r = idx >> 7, n = idx & 127;
    crS[r][n] = expf(crS[r][n] - mxS[r]);
  }
  __syncthreads();
  if (tid < Rr) {
    float s = 0.f;
    for (int n = 0; n < Nn; n++) s += crS[tid][n];
    smS[tid] = 1.0f / s;
  }
  __syncthreads();
  for (int p = 0; p < 2; p++) {        // wr = pi0*bwd + pi1*cr + pi2*fwd
    int idx = tid + p * 256;
    int r = idx >> 7, n = idx & 127;
    crS[r][n] = piS[r][0] * bwdS[r][n] + piS[r][1] * (crS[r][n] * smS[r])
              + piS[r][2] * fwdS[r][n];
  }
  __syncthreads();

  // ---- Phase J: reads[r][m] = sum_n wr[r][n] * mem_new[n][m] ----
  {
    int r = tid >> 6, m = tid & 63;
    float acc = 0.f;
    for (int n = 0; n < Nn; n++) acc += crS[r][n] * memS[n][m];
    reads_out[(size_t)b * (Rr * Mm) + tid] = acc;
  }
}

// ---------------------------------------------------------------------------
extern "C" void kernel_launch(void* const* d_in, const int* in_sizes, int n_in,
                              void* d_out, int out_size, void* d_ws, size_t ws_size,
                              hipStream_t stream) {
  const int B = 4096, D = 256, H = 512, R = 4, M = 64, OUT = 256;
  const int HEAD = 471;

  const float* x    = (const float*)d_in[0];
  const float* h    = (const float*)d_in[1];
  const float* c    = (const float*)d_in[2];
  const float* mem  = (const float*)d_in[3];
  const float* usg  = (const float*)d_in[4];
  const float* lnk  = (const float*)d_in[5];
  const float* prc  = (const float*)d_in[6];
  const float* prw  = (const float*)d_in[7];
  const float* pww  = (const float*)d_in[8];
  const float* W_ih = (const float*)d_in[9];
  const float* W_hh = (const float*)d_in[10];
  const float* b_ih = (const float*)d_in[11];
  const float* b_hh = (const float*)d_in[12];
  const float* W_out= (const float*)d_in[13];
  const float* b_out= (const float*)d_in[14];
  const float* W_hd = (const float*)d_in[15];
  const float* b_hd = (const float*)d_in[16];
  const float* W_rd = (const float*)d_in[17];
  const float* b_rd = (const float*)d_in[18];
  float* out = (float*)d_out;

  char* ws = (char*)d_ws;
  float* gates = (float*)(ws);                               // B x 4H
  float* nh    = (float*)(ws + (size_t)B * 4 * H * 4);       // B x H
  float* ctrl  = (float*)(ws + (size_t)B * 4 * H * 4
                             + (size_t)B * H * 4);           // B x 471
  float* reads = (float*)(ws + (size_t)B * 4 * H * 4
                             + (size_t)B * H * 4
                             + (size_t)B * HEAD * 4);        // B x R*M

  // 1) gates = x @ W_ih^T + (b_ih + b_hh)
  gemm_xwT_wmma<<<dim3(4 * H / 64, B / 64), 256, 0, stream>>>(
      x, W_ih, b_ih, b_hh, gates, B, 4 * H, D, 4 * H, 0);
  // 2) gates += h @ W_hh^T
  gemm_xwT_wmma<<<dim3(4 * H / 64, B / 64), 256, 0, stream>>>(
      h, W_hh, nullptr, nullptr, gates, B, 4 * H, H, 4 * H, 1);
  // 3) LSTM pointwise -> nh
  lstm_pointwise<<<(B * H + 255) / 256, 256, 0, stream>>>(gates, c, nh, B * H, H);
  // 4) ctrl = nh @ W_head^T + b_head  (N = 471, ragged)
  gemm_xwT_wmma<<<dim3((HEAD + 63) / 64, B / 64), 256, 0, stream>>>(
      nh, W_hd, b_hd, nullptr, ctrl, B, HEAD, H, HEAD, 0);
  // 5) out = nh @ W_out^T + b_out
  gemm_xwT_wmma<<<dim3(OUT / 64, B / 64), 256, 0, stream>>>(
      nh, W_out, b_out, nullptr, out, B, OUT, H, OUT, 0);
  // 6) fused DNC tail -> reads
  dnc_tail<<<B, 256, 0, stream>>>(ctrl, mem, usg, lnk, prc, prw, pww, reads);
  // 7) out += reads @ W_rd^T + b_rd
  gemm_xwT_wmma<<<dim3(OUT / 64, B / 64), 256, 0, stream>>>(
      reads, W_rd, b_rd, nullptr, out, B, OUT, R * M, OUT, 1);

  (void)in_sizes; (void)n_in; (void)out_size; (void)ws_size;
}